// HyperRuleLayer_59330678227222
// MI455X (gfx1250) — compile-verified
//
#include <hip/hip_runtime.h>
#include <math.h>

#define H  256
#define TM 32

typedef float v2f __attribute__((ext_vector_type(2)));
typedef float v8f __attribute__((ext_vector_type(8)));

__device__ __forceinline__ v8f wmma_f32x4(v2f a, v2f b, v8f c) {
  // D = A(16x4,f32) * B(4x16,f32) + C(16x16,f32)
  return __builtin_amdgcn_wmma_f32_16x16x4_f32(false, a, false, b, (short)0, c,
                                               false, false);
}

// Async global->LDS copy of 16 bytes per lane (ASYNCcnt-tracked).
__device__ __forceinline__ void async_ld_b128(unsigned lds_off, const void* g) {
  asm volatile("global_load_async_to_lds_b128 %0, %1, off"
               :: "v"(lds_off), "v"((unsigned long long)g) : "memory");
}
__device__ __forceinline__ void wait_async() {
  asm volatile("s_wait_asynccnt 0x0" ::: "memory");
}
__device__ __forceinline__ unsigned lds_off_of(const void* p) {
  // LDS flat aperture keeps the byte offset in addr[31:0]
  return (unsigned)(unsigned long long)p;
}

// One wave computes a 32 x 64 slab (2 M-tiles x 4 N-tiles) of a 32 x 256
// block:  C[32,64] += At[32,K] * W[K, n0:n0+64]
// Wp is W packed in row pairs: Wp[k/2][n] = {W[k][n], W[k+1][n]}
template <int K, int LDA>
__device__ __forceinline__ void wmma_slab32(const float* __restrict__ At,
                                            const float2* __restrict__ Wp,
                                            int lane, int wave, v8f c[2][4]) {
  const int row  = lane & 15;
  const int half = lane >> 4;
  const int n0   = wave * 64 + row;
  for (int k = 0; k < K; k += 4) {
    const int kk = k + 2 * half;
    v2f a0, a1;
    a0.x = At[row * LDA + kk];
    a0.y = At[row * LDA + kk + 1];
    a1.x = At[(row + 16) * LDA + kk];
    a1.y = At[(row + 16) * LDA + kk + 1];
    const float2* Wb = Wp + (size_t)(kk >> 1) * H;
    if (k + 8 < K) __builtin_prefetch((const void*)(Wb + 2 * H + n0), 0, 1);
#pragma unroll
    for (int tix = 0; tix < 4; ++tix) {
      const float2 bv = Wb[n0 + tix * 16];
      v2f b; b.x = bv.x; b.y = bv.y;
      c[0][tix] = wmma_f32x4(a0, b, c[0][tix]);   // rows 0..15
      c[1][tix] = wmma_f32x4(a1, b, c[1][tix]);   // rows 16..31 (B reused)
    }
  }
}

// ---------------- weight repack: Wp[k/2][n] = {W[k][n],W[k+1][n]} ---------
__global__ void pack_w_kernel(const float* __restrict__ W,
                              float2* __restrict__ Wp, int K) {
  int idx = blockIdx.x * blockDim.x + threadIdx.x;   // over (K/2)*H
  if (idx < (K / 2) * H) {
    int k2 = idx / H, n = idx % H;
    float2 p;
    p.x = W[(size_t)(2 * k2) * H + n];
    p.y = W[(size_t)(2 * k2 + 1) * H + n];
    Wp[idx] = p;
  }
}

// ---------------- map build ----------------
__global__ void fill_map_kernel(int* __restrict__ map, int R) {
  int r = blockIdx.x * blockDim.x + threadIdx.x;
  if (r < R) map[r] = -1;
}
__global__ void scatter_map_kernel(const int* __restrict__ he_tgt,
                                   int* __restrict__ map, int E) {
  int e = blockIdx.x * blockDim.x + threadIdx.x;
  if (e < E) map[he_tgt[e]] = e;
}

// ---------------- K1: segment mean + msg GEMM ----------------
__global__ __launch_bounds__(128) void seg_mean_msg_kernel(
    const float* __restrict__ x, const int* __restrict__ he_ptr,
    const int* __restrict__ he_src, const float2* __restrict__ WpM,
    const float* __restrict__ b_msg, float* __restrict__ msgbuf, int E) {
  __shared__ float At[TM][H + 4];                 // stride 260
  const int e0 = blockIdx.x * TM;
  const int t  = threadIdx.x;                     // cols 2t, 2t+1

  for (int i = 0; i < TM; ++i) {
    const int e = e0 + i;                         // uniform across block
    float ax = 0.f, ay = 0.f, inv = 0.f;
    if (e < E) {
      const int p0 = he_ptr[e], p1 = he_ptr[e + 1];
      for (int j = p0; j < p1; ++j) {
        const float2 v = ((const float2*)(x + (size_t)he_src[j] * H))[t];
        ax += v.x; ay += v.y;
      }
      inv = 1.0f / (float)(p1 - p0);
    }
    At[i][2 * t]     = ax * inv;
    At[i][2 * t + 1] = ay * inv;
  }
  __syncthreads();

  const int lane = threadIdx.x & 31, wave = threadIdx.x >> 5;
  v8f c[2][4] = {};
  wmma_slab32<H, H + 4>(&At[0][0], WpM, lane, wave, c);

  const int mrow = 8 * (lane >> 4), nl = lane & 15;
#pragma unroll
  for (int tix = 0; tix < 4; ++tix) {
    const int n = wave * 64 + tix * 16 + nl;
    const float bn = b_msg[n];
#pragma unroll
    for (int mt = 0; mt < 2; ++mt) {
#pragma unroll
      for (int v = 0; v < 8; ++v) {
        const int e = e0 + mt * 16 + mrow + v;
        if (e < E) msgbuf[(size_t)e * H + n] = c[mt][tix][v] + bn;
      }
    }
  }
}

// ---------------- K2: gate GEMM (K=512) + gated residual -----------------
__global__ __launch_bounds__(128) void gate_upd_kernel(
    const float* __restrict__ x, const int* __restrict__ he_tgt,
    const float* __restrict__ he_w, const float2* __restrict__ WpG,
    const float* __restrict__ b_gate, float* __restrict__ msgbuf, int E) {
  __shared__ float At[TM][2 * H + 4];             // [tgt_x | msg], stride 516
  __shared__ float wv[TM];
  const int e0 = blockIdx.x * TM;
  const int t  = threadIdx.x;

  // async fill: lane t covers 16B at column (t&63)*4 of half (t>>6)
  const int halfsel = t >> 6;
  const int cidx    = (t & 63) * 4;
  for (int i = 0; i < TM; ++i) {
    const int e  = e0 + i;
    const int ec = (e < E) ? e : (E - 1);
    const float* src = halfsel ? (msgbuf + (size_t)ec * H + cidx)
                               : (x + (size_t)he_tgt[ec] * H + cidx);
    async_ld_b128(lds_off_of(&At[i][halfsel * H + cidx]), src);
  }
  if (t < TM) wv[t] = (e0 + t < E) ? he_w[e0 + t] : 0.f;
  wait_async();
  __syncthreads();

  const int lane = threadIdx.x & 31, wave = threadIdx.x >> 5;
  v8f c[2][4] = {};
  wmma_slab32<2 * H, 2 * H + 4>(&At[0][0], WpG, lane, wave, c);

  const int mrow = 8 * (lane >> 4), nl = lane & 15;
#pragma unroll
  for (int tix = 0; tix < 4; ++tix) {
    const int n = wave * 64 + tix * 16 + nl;
    const float bg = b_gate[n];
#pragma unroll
    for (int mt = 0; mt < 2; ++mt) {
#pragma unroll
      for (int v = 0; v < 8; ++v) {
        const int m = mt * 16 + mrow + v;
        const int e = e0 + m;
        const float z  = c[mt][tix][v] + bg;
        const float g  = 1.0f / (1.0f + __expf(-z));     // sigmoid
        const float tg = At[m][n];
        const float ms = At[m][H + n];
        const float u  = tg + wv[m] * g * ms;            // gated residual
        if (e < E) msgbuf[(size_t)e * H + n] = u;        // upd over msg
      }
    }
  }
}

// ---------------- K3: final GEMM over all R rows with row-select ---------
__global__ __launch_bounds__(128) void final_kernel(
    const float* __restrict__ x, const float* __restrict__ updbuf,
    const int* __restrict__ map, const float2* __restrict__ WpU,
    const float* __restrict__ b_upd, float* __restrict__ out, int R) {
  __shared__ float At[TM][H + 4];
  const int r0 = blockIdx.x * TM;
  const int t  = threadIdx.x;

  // async fill: 2 rows per round, 64 lanes x 16B per row
  const int rsub = t >> 6;
  const int cidx = (t & 63) * 4;
  for (int rr = 0; rr < TM / 2; ++rr) {
    const int i = rr * 2 + rsub;
    const int r = (r0 + i < R) ? (r0 + i) : (R - 1);
    const int m = map[r];
    const float* src = (m >= 0) ? (updbuf + (size_t)m * H + cidx)
                                : (x + (size_t)r * H + cidx);
    async_ld_b128(lds_off_of(&At[i][cidx]), src);
  }
  wait_async();
  __syncthreads();

  const int lane = threadIdx.x & 31, wave = threadIdx.x >> 5;
  v8f c[2][4] = {};
  wmma_slab32<H, H + 4>(&At[0][0], WpU, lane, wave, c);

  const int mrow = 8 * (lane >> 4), nl = lane & 15;
#pragma unroll
  for (int tix = 0; tix < 4; ++tix) {
    const int n = wave * 64 + tix * 16 + nl;
    const float bn = b_upd[n];
#pragma unroll
    for (int mt = 0; mt < 2; ++mt) {
#pragma unroll
      for (int v = 0; v < 8; ++v) {
        const int r = r0 + mt * 16 + mrow + v;
        const float o = fminf(fmaxf(c[mt][tix][v] + bn, 0.0f), 1.0f);
        if (r < R) out[(size_t)r * H + n] = o;
      }
    }
  }
}

extern "C" void kernel_launch(void* const* d_in, const int* in_sizes, int n_in,
                              void* d_out, int out_size, void* d_ws,
                              size_t ws_size, hipStream_t stream) {
  const float* x      = (const float*)d_in[0];
  const int*   he_ptr = (const int*)d_in[1];
  const int*   he_src = (const int*)d_in[2];
  const int*   he_tgt = (const int*)d_in[3];
  const float* he_w   = (const float*)d_in[4];
  const float* W_msg  = (const float*)d_in[5];
  const float* b_msg  = (const float*)d_in[6];
  const float* W_gate = (const float*)d_in[7];
  const float* b_gate = (const float*)d_in[8];
  const float* W_upd  = (const float*)d_in[9];
  const float* b_upd  = (const float*)d_in[10];
  float* out = (float*)d_out;

  const int E = in_sizes[3];          // he_tgt count
  const int R = in_sizes[0] / H;      // rows of x

  // workspace: [msg/upd: E*H f32][map: R i32][WpM][WpG][WpU]
  char* ws = (char*)d_ws;
  float* msgbuf = (float*)ws;
  size_t off = (size_t)E * H * sizeof(float);
  int* map = (int*)(ws + off);
  off += (size_t)R * sizeof(int);
  off = (off + 15) & ~(size_t)15;
  float2* WpM = (float2*)(ws + off); off += (size_t)(H / 2) * H * sizeof(float2);
  float2* WpG = (float2*)(ws + off); off += (size_t)H * H * sizeof(float2);
  float2* WpU = (float2*)(ws + off);

  pack_w_kernel<<<((H / 2) * H + 255) / 256, 256, 0, stream>>>(W_msg, WpM, H);
  pack_w_kernel<<<(H * H + 255) / 256, 256, 0, stream>>>(W_gate, WpG, 2 * H);
  pack_w_kernel<<<((H / 2) * H + 255) / 256, 256, 0, stream>>>(W_upd, WpU, H);
  fill_map_kernel<<<(R + 255) / 256, 256, 0, stream>>>(map, R);
  scatter_map_kernel<<<(E + 255) / 256, 256, 0, stream>>>(he_tgt, map, E);

  seg_mean_msg_kernel<<<(E + TM - 1) / TM, 128, 0, stream>>>(
      x, he_ptr, he_src, WpM, b_msg, msgbuf, E);
  gate_upd_kernel<<<(E + TM - 1) / TM, 128, 0, stream>>>(
      x, he_tgt, he_w, WpG, b_gate, msgbuf, E);
  final_kernel<<<(R + TM - 1) / TM, 128, 0, stream>>>(
      x, msgbuf, map, WpU, b_upd, out, R);
}